// SimpleConv_90151363543629
// MI455X (gfx1250) — compile-verified
//
#include <hip/hip_runtime.h>
#include <stdint.h>

#define TILE 64
#define HALO 5
#define IN   (TILE + 2*HALO)   // 74
#define ST   IN                // LDS row stride = 74 floats (TDM writes contiguous rows)
#define NT   256
#define IMGW 1024
#define IMGH 1024

typedef __attribute__((address_space(1))) int g1_int;   // clang prints as "__device__ int *"
typedef __attribute__((address_space(3))) int l3_int;   // "__shared__ int *"

typedef uint32_t u32x4 __attribute__((ext_vector_type(4)));
typedef int32_t  i32x8 __attribute__((ext_vector_type(8)));
typedef int32_t  i32x4 __attribute__((ext_vector_type(4)));

#if __has_builtin(__builtin_amdgcn_tensor_load_to_lds)
  #define HAVE_TDM 1
#else
  #define HAVE_TDM 0
#endif

// ---- CDNA5 async global -> LDS copy (per-lane, tracked by ASYNCcnt) --------
__device__ __forceinline__ void async_copy_f32(const float* gsrc, float* ldst) {
#if __has_builtin(__builtin_amdgcn_global_load_async_to_lds_b32)
    __builtin_amdgcn_global_load_async_to_lds_b32(
        (g1_int*)(uintptr_t)gsrc,
        (l3_int*)(uint32_t)(uintptr_t)ldst,
        0, 0);
#else
    uint32_t l = (uint32_t)(uintptr_t)ldst;
    asm volatile("global_load_async_to_lds_b32 %0, %1, off"
                 :: "v"(l), "v"(gsrc) : "memory");
#endif
}

__device__ __forceinline__ void wait_async0() {
#if __has_builtin(__builtin_amdgcn_s_wait_asynccnt)
    __builtin_amdgcn_s_wait_asynccnt(0);
#else
    asm volatile("s_wait_asynccnt 0" ::: "memory");
#endif
}

__device__ __forceinline__ void wait_tensor0() {
#if __has_builtin(__builtin_amdgcn_s_wait_tensorcnt)
    __builtin_amdgcn_s_wait_tensorcnt(0);
#else
    asm volatile("s_wait_tensorcnt 0" ::: "memory");
#endif
}

// ---- TDM: one DMA of a 74x74 f32 tile (stride 1024) into LDS ---------------
// D# per CDNA5 ISA: group0 = {count|flags, lds_addr,
// global_addr[56:0], type=2}; group1 = {data_size=4B, tensor_dim0/1=1024,
// tile_dim0/1=74, tensor_dim0_stride=1024}; groups 2/3 unused (2-D).
__device__ __forceinline__ void tdm_load_tile(const float* gsrc, uint32_t lds_addr) {
#if HAVE_TDM
    const uint64_t ga = (uint64_t)(uintptr_t)gsrc;
    u32x4 g0;
    g0[0] = 1u;                                              // count=1, user D#
    g0[1] = lds_addr;                                        // lds_addr (bytes)
    g0[2] = (uint32_t)ga;                                    // global_addr[31:0]
    g0[3] = (uint32_t)((ga >> 32) & 0x01FFFFFFu)             // global_addr[56:32]
          | (2u << 30);                                      // type = 2 ("image")
    i32x8 g1;
    g1[0] = (int)(2u << 16);                                 // data_size=2 (4B), no pad/iter/mcast
    g1[1] = (int)((IMGW & 0xFFFFu) << 16);                   // tensor_dim0[15:0] @ bits 63:48
    g1[2] = (int)(((IMGW >> 16) & 0xFFFFu)                   // tensor_dim0[31:16]
          | ((IMGH & 0xFFFFu) << 16));                       // tensor_dim1[15:0]
    g1[3] = (int)(((IMGH >> 16) & 0xFFFFu)                   // tensor_dim1[31:16]
          | ((uint32_t)IN << 16));                           // tile_dim0 = 74
    g1[4] = (int)IN;                                         // tile_dim1 = 74, tile_dim2 = 0
    g1[5] = IMGW;                                            // tensor_dim0_stride[31:0]
    g1[6] = 0;                                               // stride0 hi | dim1_stride[15:0]
    g1[7] = (int)((uint32_t)(IMGW * IMGH) >> 16);            // tensor_dim1_stride[47:16]
    i32x4 z4 = {0, 0, 0, 0};
    i32x8 z8 = {0, 0, 0, 0, 0, 0, 0, 0};
    __builtin_amdgcn_tensor_load_to_lds(g0, g1, z4, z4, z8, 0);
#else
    (void)gsrc; (void)lds_addr;
#endif
}

// ---- One separable 3x3 Gaussian stage, entirely in LDS ----------------------
// S -> D over the shrinking valid region [s, 74-s)^2. Per task: one column,
// run of CH rows, vertical sliding window of horizontal sums
// h(y,x) = S[y][x-1] + 2 S[y][x] + S[y][x+1]; out = (h0+2h1+h2)/16.
// Reference zero-pads at every conv, so out-of-image pixels are forced to 0.
__device__ __forceinline__ void stage_pass(const float* __restrict__ S,
                                           float* __restrict__ D,
                                           int s, bool relu,
                                           int gx0, int gy0, int tid) {
    const int c0 = s, c1 = IN - s;
    const int C  = c1 - c0;              // even for all stages
    const int r0 = s, r1 = IN - s;
    const int R  = r1 - r0;
    const int CH = 16;
    const int nch   = (R + CH - 1) / CH;
    const int total = C * nch;
    for (int t = tid; t < total; t += NT) {
        const int x  = c0 + (t % C);
        const int yc = t / C;
        const int y  = r0 + yc * CH;
        const int ye = (y + CH < r1) ? (y + CH) : r1;
        const bool okx = ((unsigned)(gx0 + x) < (unsigned)IMGW);
        const float* p = S + (y - 1) * ST + (x - 1);
        float h0 = p[0] + 2.0f * p[1] + p[2];  p += ST;   // row y-1
        float h1 = p[0] + 2.0f * p[1] + p[2];  p += ST;   // row y
        for (int yy = y; yy < ye; ++yy) {
            float h2 = p[0] + 2.0f * p[1] + p[2];  p += ST;  // row yy+1
            float v  = (h0 + 2.0f * h1 + h2) * 0.0625f;
            if (relu) v = fmaxf(v, 0.0f);
            const bool ok = okx && ((unsigned)(gy0 + yy) < (unsigned)IMGH);
            D[yy * ST + x] = ok ? v : 0.0f;
            h0 = h1; h1 = h2;
        }
    }
}

// ---- Fused 5-stage depthwise Gaussian chain --------------------------------
__global__ __launch_bounds__(NT)
void fused_gauss5_kernel(const float* __restrict__ x, float* __restrict__ out) {
    __shared__ float A[IN * ST];
    __shared__ float B[IN * ST];

    const int tid = threadIdx.x;
    const int tx  = blockIdx.x;
    const int ty  = blockIdx.y;
    const int img = blockIdx.z;          // n*3 + c (depthwise: channels independent)

    const int gx0 = tx * TILE - HALO;
    const int gy0 = ty * TILE - HALO;
    const float* src = x + (size_t)img * ((size_t)IMGW * IMGH);

    const bool border = (tx == 0) | (ty == 0) |
                        (tx == (int)gridDim.x - 1) | (ty == (int)gridDim.y - 1);

    if (HAVE_TDM && !border) {
        // Interior tile: one TDM DMA of the whole 74x74 tile, issued by wave 0.
        if (tid < 32) {
            tdm_load_tile(src + (size_t)gy0 * IMGW + gx0,
                          (uint32_t)(uintptr_t)&A[0]);
            wait_tensor0();
        }
    } else {
        // Border tile (or no TDM): zero-pad then per-lane async loads.
        if (border) {
            for (int i = tid; i < IN * ST; i += NT) A[i] = 0.0f;
            __syncthreads();             // ds writes land before async writes
        }
        for (int i = tid; i < IN * IN; i += NT) {
            const int r  = i / IN;
            const int c  = i - r * IN;
            const int gy = gy0 + r;
            const int gx = gx0 + c;
            if ((unsigned)gy < (unsigned)IMGH && (unsigned)gx < (unsigned)IMGW)
                async_copy_f32(src + (size_t)gy * IMGW + gx, &A[r * ST + c]);
        }
        wait_async0();
    }
    __syncthreads();

    stage_pass(A, B, 1, false, gx0, gy0, tid); __syncthreads();
    stage_pass(B, A, 2, true , gx0, gy0, tid); __syncthreads();  // ReLU after conv2
    stage_pass(A, B, 3, false, gx0, gy0, tid); __syncthreads();
    stage_pass(B, A, 4, false, gx0, gy0, tid); __syncthreads();
    stage_pass(A, B, 5, false, gx0, gy0, tid); __syncthreads();

    // Write the 64x64 result (buffer region [5,69)^2) with coalesced float4 stores.
    float* dst = out + (size_t)img * ((size_t)IMGW * IMGH)
                     + (size_t)(ty * TILE) * IMGW + tx * TILE;
    const int xq = (tid & 15) * 4;       // 0..60
    const int yb = tid >> 4;             // 0..15
    for (int k = 0; k < 4; ++k) {
        const int y = yb + k * 16;       // 0..63
        const float* b = &B[(HALO + y) * ST + HALO + xq];
        float4 v = make_float4(b[0], b[1], b[2], b[3]);
        *(float4*)(dst + (size_t)y * IMGW + xq) = v;
    }
}

extern "C" void kernel_launch(void* const* d_in, const int* in_sizes, int n_in,
                              void* d_out, int out_size, void* d_ws, size_t ws_size,
                              hipStream_t stream) {
    (void)n_in; (void)d_ws; (void)ws_size; (void)out_size;
    const float* x   = (const float*)d_in[0];
    float*       out = (float*)d_out;
    const int imgs = in_sizes[0] / (IMGW * IMGH);      // 16*3 = 48
    dim3 grid(IMGW / TILE, IMGH / TILE, imgs);         // (16,16,48)
    fused_gauss5_kernel<<<grid, NT, 0, stream>>>(x, out);
}